// SingleVQ_66322884984997
// MI455X (gfx1250) — compile-verified
//
#include <hip/hip_runtime.h>
#include <hip/hip_bf16.h>

typedef __attribute__((ext_vector_type(2))) float v2f;
typedef __attribute__((ext_vector_type(8))) float v8f;

#define EMBED     32
#define KCODES    32768
#define NTOK      4096          // 4 * 32 * 32 tokens
#define SEGS      8
#define SEG_CODES (KCODES / SEGS)   // 4096 codes per segment
#define CH_STRIDE 1024          // H*W elements between channels in z layout
#define B_STRIDE  32768         // C*H*W elements between batches
#define LOSS_SCALE (1.25f / 131072.0f)   // (1+BETA)/numel

// ---------------------------------------------------------------------------
// Kernel 0: precompute ||e_k||^2 for every code (one thread per code).
// ---------------------------------------------------------------------------
__global__ __launch_bounds__(256) void vq_norms_kernel(
    const float* __restrict__ cb, float* __restrict__ norms)
{
    const int k = blockIdx.x * blockDim.x + threadIdx.x;      // 0..32767
    const float4* row = (const float4*)(cb + (size_t)k * EMBED);
    float s = 0.0f;
#pragma unroll
    for (int i = 0; i < 8; ++i) {
        const float4 v = row[i];
        s += v.x * v.x + v.y * v.y + v.z * v.z + v.w * v.w;
    }
    norms[k] = s;
}

// ---------------------------------------------------------------------------
// Kernel 1: WMMA fp32 distance sweep + per-segment argmin.
// One wave handles one 16-token tile x one 4096-code segment.
// score = ||e||^2 - 2*z.e  (||z||^2 constant per row -> dropped for argmin).
// ---------------------------------------------------------------------------
__global__ __launch_bounds__(256) void vq_argmin_kernel(
    const float* __restrict__ z,        // [4][32][32][32]
    const float* __restrict__ cb,       // [32768][32]
    const float* __restrict__ norms,    // [32768]
    float* __restrict__ seg_val,        // [NTOK][SEGS]
    int*   __restrict__ seg_idx)        // [NTOK][SEGS]
{
    const int wave = (blockIdx.x * blockDim.x + threadIdx.x) >> 5;
    const int lane = threadIdx.x & 31;
    const int tile = wave / SEGS;           // 0..255
    const int seg  = wave % SEGS;           // 0..7
    const int t0   = tile * 16;
    const int half = lane >> 4;             // 0: K%4 in {0,1}, 1: K%4 in {2,3}
    const int l16  = lane & 15;

    // ---- Load A fragments (z tile) once: token t0+l16, features 4s+2*half+{0,1}
    const int tok = t0 + l16;
    const int b   = tok >> 10;
    const int hw  = tok & 1023;
    const float* zb = z + (size_t)b * B_STRIDE + hw;

    v2f a[8];
#pragma unroll
    for (int s = 0; s < 8; ++s) {
        const int c = 4 * s + 2 * half;
        a[s][0] = zb[(size_t)c * CH_STRIDE];
        a[s][1] = zb[(size_t)(c + 1) * CH_STRIDE];
    }

    float minval[8];
    int   minidx[8];
#pragma unroll
    for (int r = 0; r < 8; ++r) { minval[r] = 3.0e38f; minidx[r] = 0; }

    const int cbase = seg * SEG_CODES;
#pragma unroll 2
    for (int c0 = cbase; c0 < cbase + SEG_CODES; c0 += 16) {
        const int code = c0 + l16;
        const float* crow = cb + (size_t)code * EMBED + 2 * half;

        // prefetch next chunk's row (same lane role) into caches
        __builtin_prefetch(crow + 16 * EMBED, 0, 3);

        // ---- Load B fragments (16 codes) + precomputed norm (1 extra b32)
        v2f bf[8];
#pragma unroll
        for (int s = 0; s < 8; ++s) {
            bf[s] = *(const v2f*)(crow + 4 * s);   // contiguous b64 load
        }
        const float nrm = norms[code];             // column N = lane&15 both halves

        // ---- 16x16 dot-product tile via 8 x V_WMMA_F32_16X16X4_F32 (K=32)
        v8f acc = {};
#pragma unroll
        for (int s = 0; s < 8; ++s) {
            acc = __builtin_amdgcn_wmma_f32_16x16x4_f32(
                /*neg_a=*/false, a[s], /*neg_b=*/false, bf[s],
                /*c_mod=*/(short)0, acc, /*reuse_a=*/false, /*reuse_b=*/false);
        }

        // ---- update running argmin per C-row slot:
        // one shared v_cmp feeding a dual v_cndmask pair (idx + val)
#pragma unroll
        for (int r = 0; r < 8; ++r) {
            const float d = __builtin_fmaf(-2.0f, acc[r], nrm);
            const bool lt = d < minval[r];
            minidx[r] = lt ? code : minidx[r];
            minval[r] = lt ? d    : minval[r];
        }
    }

    // ---- reduce (minval,minidx) across the 16 lanes of each half-wave
#pragma unroll
    for (int r = 0; r < 8; ++r) {
#pragma unroll
        for (int off = 8; off >= 1; off >>= 1) {
            const float ov = __shfl_xor(minval[r], off, 32);
            const int   oi = __shfl_xor(minidx[r], off, 32);
            if (ov < minval[r] || (ov == minval[r] && oi < minidx[r])) {
                minval[r] = ov; minidx[r] = oi;
            }
        }
    }

    // lane 0 holds tokens t0+0..7 (C rows M=0..7), lane 16 holds t0+8..15
    if (l16 == 0) {
#pragma unroll
        for (int r = 0; r < 8; ++r) {
            const int tw = t0 + half * 8 + r;
            seg_val[(size_t)tw * SEGS + seg] = minval[r];
            seg_idx[(size_t)tw * SEGS + seg] = minidx[r];
        }
    }
}

// ---------------------------------------------------------------------------
// Kernel 2: reduce segment winners, gather zq into [B,C,H,W], accumulate loss.
// One thread per token.
// ---------------------------------------------------------------------------
__global__ __launch_bounds__(256) void vq_gather_kernel(
    const float* __restrict__ z,
    const float* __restrict__ cb,
    const float* __restrict__ seg_val,
    const int*   __restrict__ seg_idx,
    float* __restrict__ out)            // [131072] zq, out[131072] = loss
{
    const int tok = blockIdx.x * blockDim.x + threadIdx.x;   // 0..4095

    float bestv = 3.0e38f;
    int   besti = 0;
#pragma unroll
    for (int s = 0; s < SEGS; ++s) {
        const float v = seg_val[(size_t)tok * SEGS + s];
        const int   i = seg_idx[(size_t)tok * SEGS + s];
        if (v < bestv || (v == bestv && i < besti)) { bestv = v; besti = i; }
    }

    const int b  = tok >> 10;
    const int hw = tok & 1023;
    const float* crow = cb + (size_t)besti * EMBED;
    const float* zb   = z   + (size_t)b * B_STRIDE + hw;
    float*       ob   = out + (size_t)b * B_STRIDE + hw;

    float sq = 0.0f;
#pragma unroll
    for (int c = 0; c < EMBED; ++c) {
        const float e = crow[c];
        const float d = e - zb[(size_t)c * CH_STRIDE];
        sq += d * d;
        ob[(size_t)c * CH_STRIDE] = e;       // zq_st forward value == zq
    }

    // block reduction of squared error, then one atomic per block
    __shared__ float red[256];
    red[threadIdx.x] = sq;
    __syncthreads();
    for (int s = 128; s > 0; s >>= 1) {
        if (threadIdx.x < s) red[threadIdx.x] += red[threadIdx.x + s];
        __syncthreads();
    }
    if (threadIdx.x == 0) atomicAdd(out + 131072, red[0] * LOSS_SCALE);
}

// ---------------------------------------------------------------------------
extern "C" void kernel_launch(void* const* d_in, const int* in_sizes, int n_in,
                              void* d_out, int out_size, void* d_ws, size_t ws_size,
                              hipStream_t stream) {
    const float* z  = (const float*)d_in[0];   // 131072 fp32
    const float* cb = (const float*)d_in[1];   // 32768*32 fp32
    float* out = (float*)d_out;                // 131072 zq + 1 loss

    float* seg_val = (float*)d_ws;                       // NTOK*SEGS floats
    int*   seg_idx = (int*)(seg_val + NTOK * SEGS);      // NTOK*SEGS ints
    float* norms   = (float*)(seg_idx + NTOK * SEGS);    // KCODES floats

    // zero the loss accumulator (d_out is poisoned before timing)
    hipMemsetAsync(out + 131072, 0, sizeof(float), stream);

    // 32768 code norms, 1 thread each
    vq_norms_kernel<<<KCODES / 256, 256, 0, stream>>>(cb, norms);

    // 2048 waves = 256 blocks x 256 threads (8 waves: tile=blockIdx, seg=warp)
    vq_argmin_kernel<<<256, 256, 0, stream>>>(z, cb, norms, seg_val, seg_idx);

    // 4096 tokens, 1 thread each
    vq_gather_kernel<<<NTOK / 256, 256, 0, stream>>>(z, cb, seg_val, seg_idx, out);
}